// MaskLoss_73134703116963
// MI455X (gfx1250) — compile-verified
//
#include <hip/hip_runtime.h>
#include <hip/hip_bf16.h>

// MaskLoss for MI455X (gfx1250, wave32).
// Pipeline: [iou/argmax 512x100] -> [per-detection 28x28 bilinear crop + |diff|]
// -> [scalar finalize]. Wave-level sum reduction done with V_WMMA_F32_16X16X4_F32
// (ones-matrix trick) to use the CDNA5 matrix pipe at full f32 precision.

typedef float v2f __attribute__((ext_vector_type(2)));
typedef float v8f __attribute__((ext_vector_type(8)));

#define NDET 512
#define KANN 100
#define MH   28
#define MW   28
#define IMH  800
#define IMW  800
#define IOU_TH 0.5f
#define EPSF 1e-7f

// ---------------- Kernel A: IoU + argmax + valid count, init accumulators ----
__global__ __launch_bounds__(NDET)
void iou_argmax_kernel(const float* __restrict__ det,   // (512,4)
                       const float* __restrict__ ann,   // (100,4)
                       int*   __restrict__ inds,        // (512,)
                       float* __restrict__ validf,      // (512,)
                       float* __restrict__ acc,         // scalar
                       float* __restrict__ nsel)        // scalar
{
    __shared__ float sann[KANN * 4];
    __shared__ float red[NDET];
    const int t = threadIdx.x;

    if (t < KANN * 4) sann[t] = ann[t];
    __syncthreads();

    const float x1 = det[t * 4 + 0];
    const float y1 = det[t * 4 + 1];
    const float x2 = det[t * 4 + 2];
    const float y2 = det[t * 4 + 3];
    const float areaA = (x2 - x1) * (y2 - y1);

    float best = -1.0f;
    int   bi   = 0;
#pragma unroll 4
    for (int k = 0; k < KANN; ++k) {
        const float ax1 = sann[k * 4 + 0];
        const float ay1 = sann[k * 4 + 1];
        const float ax2 = sann[k * 4 + 2];
        const float ay2 = sann[k * 4 + 3];
        const float areaB = (ax2 - ax1) * (ay2 - ay1);
        float iw = fminf(x2, ax2) - fmaxf(x1, ax1);
        float ih = fminf(y2, ay2) - fmaxf(y1, ay1);
        iw = fmaxf(iw, 0.0f);
        ih = fmaxf(ih, 0.0f);
        const float inter = iw * ih;
        const float ua    = fmaxf(areaA + areaB - inter, EPSF);
        const float iou   = inter / ua;
        if (iou > best) { best = iou; bi = k; }   // strict '>' keeps first max (jnp.argmax)
    }

    const float vld = (best >= IOU_TH) ? 1.0f : 0.0f;
    inds[t]   = bi;
    validf[t] = vld;

    red[t] = vld;
    __syncthreads();
    for (int s = NDET / 2; s > 0; s >>= 1) {
        if (t < s) red[t] += red[t + s];
        __syncthreads();
    }
    if (t == 0) {
        *nsel = red[0];
        *acc  = 0.0f;   // init global accumulator (ws is poisoned by harness)
    }
}

// ---------------- Kernel B: bilinear crop + |diff|, WMMA wave reduction -------
__global__ __launch_bounds__(256)
void mask_diff_kernel(const float* __restrict__ det,    // (512,4)
                      const float* __restrict__ masks,  // (512,28,28,100)
                      const float* __restrict__ mt,     // (100,800,800)
                      const int*   __restrict__ inds,
                      const float* __restrict__ validf,
                      float* __restrict__ acc)
{
    const int n = blockIdx.x;
    if (validf[n] == 0.0f) return;            // block-uniform: contribution is 0

    const int k = inds[n];
    const float* __restrict__ img = mt + (size_t)k * (IMH * IMW);

    // boxes = [y1/H, x1/W, y2/H, x2/W]
    const float bx1 = det[n * 4 + 0] * (1.0f / IMW);
    const float by1 = det[n * 4 + 1] * (1.0f / IMH);
    const float bx2 = det[n * 4 + 2] * (1.0f / IMW);
    const float by2 = det[n * 4 + 3] * (1.0f / IMH);

    const float gscale = 1.0f / (float)(MH - 1);   // 1/27

    float p = 0.0f;
    for (int idx = threadIdx.x; idx < MH * MW; idx += blockDim.x) {
        const int i = idx / MW;
        const int j = idx - i * MW;

        const float gy   = (float)i * gscale;
        const float gx   = (float)j * gscale;
        const float in_y = (by1 + (by2 - by1) * gy) * (float)(IMH - 1);
        const float in_x = (bx1 + (bx2 - bx1) * gx) * (float)(IMW - 1);

        const bool vy = (in_y >= 0.0f) && (in_y <= (float)(IMH - 1));
        const bool vx = (in_x >= 0.0f) && (in_x <= (float)(IMW - 1));

        const float y0f = floorf(in_y);
        const float x0f = floorf(in_x);
        const float yf  = in_y - y0f;
        const float xf  = in_x - x0f;

        int y0i = (int)y0f; y0i = min(max(y0i, 0), IMH - 1);
        int x0i = (int)x0f; x0i = min(max(x0i, 0), IMW - 1);
        const int y1i = min(y0i + 1, IMH - 1);
        const int x1i = min(x0i + 1, IMW - 1);

        const float v00 = img[(size_t)y0i * IMW + x0i];
        const float v01 = img[(size_t)y0i * IMW + x1i];
        const float v10 = img[(size_t)y1i * IMW + x0i];
        const float v11 = img[(size_t)y1i * IMW + x1i];

        const float top = v00 + (v01 - v00) * xf;
        const float bot = v10 + (v11 - v10) * xf;
        float out = top + (bot - top) * yf;
        out = (vy && vx) ? out : 0.0f;

        const float msel = masks[(((size_t)n * MH + i) * MW + j) * KANN + k];
        p += fabsf(msel - out);
    }

    // ---- wave32 sum via V_WMMA_F32_16X16X4_F32 with an all-ones B matrix ----
    // A (16x4 f32) stripe: lane L<16 holds (K=0,K=1) of row M=L; lane L>=16 holds
    // (K=2,K=3) of row M=L-16. Put each lane's partial in slot 0 -> D row sums
    // equal pairwise lane sums; summing a lane's 8 D-VGPRs + one cross-half
    // shuffle gives the exact f32 wave total.
    v2f a; a.x = p;     a.y = 0.0f;
    v2f b; b.x = 1.0f;  b.y = 1.0f;
    v8f c = {};
    c = __builtin_amdgcn_wmma_f32_16x16x4_f32(
            /*neg_a=*/false, a, /*neg_b=*/false, b,
            /*c_mod=*/(short)0, c, /*reuse_a=*/false, /*reuse_b=*/false);

    float s = c[0] + c[1] + c[2] + c[3] + c[4] + c[5] + c[6] + c[7];
    // lanes 0..15 hold sum(rows 0..7); lanes 16..31 hold sum(rows 8..15)
    const float tot = __shfl(s, 0, 32) + __shfl(s, 16, 32);

    if ((threadIdx.x & 31) == 0) atomicAdd(acc, tot);
}

// ---------------- Kernel C: finalize --------------------------------------
__global__ void finalize_kernel(const float* __restrict__ acc,
                                const float* __restrict__ nsel,
                                float* __restrict__ out)
{
    const float divisor = fmaxf(nsel[0] * (float)(MH * MW), 1.0f);
    out[0] = acc[0] / divisor;
}

extern "C" void kernel_launch(void* const* d_in, const int* in_sizes, int n_in,
                              void* d_out, int out_size, void* d_ws, size_t ws_size,
                              hipStream_t stream) {
    const float* det  = (const float*)d_in[0];   // (1,512,4)
    const float* msk  = (const float*)d_in[1];   // (1,512,28,28,100)
    const float* ann  = (const float*)d_in[2];   // (1,100,4)
    const float* mt   = (const float*)d_in[3];   // (1,100,800,800)
    float*       out  = (float*)d_out;

    // workspace layout: acc | nsel | inds[512] | validf[512]
    float* acc    = (float*)d_ws;
    float* nsel   = acc + 1;
    int*   inds   = (int*)(acc + 2);
    float* validf = (float*)(inds + NDET);

    iou_argmax_kernel<<<1, NDET, 0, stream>>>(det, ann, inds, validf, acc, nsel);
    mask_diff_kernel<<<NDET, 256, 0, stream>>>(det, msk, mt, inds, validf, acc);
    finalize_kernel<<<1, 1, 0, stream>>>(acc, nsel, out);
}